// ComplexRNN_10969346474391
// MI455X (gfx1250) — compile-verified
//
#include <hip/hip_runtime.h>
#include <math.h>

// ComplexRNN on MI455X (gfx1250, wave32):
//   10-layer tanh RNN, B=1024, T=512, D=16, H=30, FC(H->1) on last step.
//
// All 10 layers FUSED into one kernel as a software-pipelined wavefront:
// block (l, bt) runs layer l for batch tile bt; it consumes layer l-1's
// rows via acquire/release progress flags (chunked every 4 timesteps).
// Dependencies point strictly to lower blockIdx (in-order prefix dispatch
// => no deadlock even under partial residency). Steady state overlaps all
// layers: ~T + 8L serial steps instead of L*T.
//
// Per layer-step: M=16 batch tile per wave, hidden padded N=32/K=32,
// f32 WMMA 16x16x4 for input GEMM + recurrence; h-state round-trips
// through LDS (D-fragment -> A-fragment); input A-fragments software-
// pipelined in registers; recurrence split into 2 independent WMMA chains;
// branch-free tanh via v_exp_f32 + v_rcp_f32.
// Activations ping-pong in d_ws as [B][T][32] f32 (L2-resident, 2x64MB).

typedef __attribute__((ext_vector_type(2))) float v2f;  // A/B frag, 16x16x4 f32
typedef __attribute__((ext_vector_type(8))) float v8f;  // C/D frag

#define BB   1024
#define TT   512
#define DD   16
#define HH   30
#define LL   10
#define NBT  (BB / 16)   // 64 batch tiles
#define OSTR 32          // padded column stride of activation buffers
#define LDSS 34          // LDS row stride (even -> aligned b64 loads)
#define CHK  4           // flag publication granularity (timesteps)

static __device__ __forceinline__ v8f wmma_f32_k4(v2f a, v2f b, v8f c) {
    return __builtin_amdgcn_wmma_f32_16x16x4_f32(false, a, false, b,
                                                 (short)0, c, false, false);
}

// Branch-free tanh: tanh(x) = sign(x) * (1 - e) / (1 + e), e = exp(-2|x|).
static __device__ __forceinline__ float tanh_fast(float x) {
    const float ax = __builtin_fabsf(x);
    const float e  = __expf(-2.0f * ax);
    const float r  = (1.0f - e) * __builtin_amdgcn_rcpf(1.0f + e);
    return __builtin_copysignf(r, x);
}

__global__ void zero_flags(int* flags, int n) {
    const int i = blockIdx.x * blockDim.x + threadIdx.x;
    if (i < n) flags[i] = 0;
}

// Fused pipelined RNN: grid = LL*NBT blocks of 32 threads (1 wave each).
__global__ __launch_bounds__(32) void rnn_pipeline(
    const float* __restrict__ x, const float* __restrict__ W_ih0,
    const float* __restrict__ W_ih, const float* __restrict__ W_hh,
    const float* __restrict__ b_ih, const float* __restrict__ b_hh,
    float* __restrict__ buf0, float* __restrict__ buf1,
    int* __restrict__ flags) {
    __shared__ float hbuf[16 * LDSS];  // h state: rows=batch(m), cols=hidden(n)

    const int  bid   = blockIdx.x;
    const int  l     = bid >> 6;  // layer (NBT=64)
    const int  bt    = bid & (NBT - 1);
    const bool first = (l == 0);
    const bool last  = (l == LL - 1);
    const int  lane  = threadIdx.x;
    const int  c     = lane >> 4;
    const int  lo    = lane & 15;

    // layer l>=1 reads bufs[(l+1)&1]; layer l writes bufs[l&1]
    const float* in   = first ? x : ((l & 1) ? buf0 : buf1);
    float*       out  = (l & 1) ? buf1 : buf0;
    const int    istr = first ? DD : OSTR;
    const int    kin  = first ? DD : HH;
    const float* Wi   = first ? W_ih0 : (W_ih + (size_t)(l - 1) * HH * HH);
    const float* Wh   = W_hh + (size_t)l * HH * HH;
    const float* bi   = b_ih + l * HH;
    const float* bh   = b_hh + l * HH;
    int*         myflag  = flags + bid;
    const int*   srcflag = flags + bid - NBT;  // producer: strictly lower bid

    // h_0 = 0
    for (int i = lane; i < 16 * LDSS; i += 32) hbuf[i] = 0.f;
    __syncthreads();

    // B-fragments of Wi^T: B[k][n] = Wi[n*kin + k]   (out = in @ Wi^T)
    v2f wiB[8][2];
#pragma unroll
    for (int kk = 0; kk < 8; ++kk)
#pragma unroll
        for (int nt = 0; nt < 2; ++nt)
#pragma unroll
            for (int r = 0; r < 2; ++r) {
                const int k = 4 * kk + 2 * c + r;
                const int n = nt * 16 + lo;
                wiB[kk][nt][r] = (k < kin && n < HH) ? Wi[n * kin + k] : 0.f;
            }

    // B-fragments of Wh^T: B[k][n] = Wh[n*HH + k]
    v2f whB[8][2];
#pragma unroll
    for (int kk = 0; kk < 8; ++kk)
#pragma unroll
        for (int nt = 0; nt < 2; ++nt)
#pragma unroll
            for (int r = 0; r < 2; ++r) {
                const int k = 4 * kk + 2 * c + r;
                const int n = nt * 16 + lo;
                whB[kk][nt][r] = (k < HH && n < HH) ? Wh[n * HH + k] : 0.f;
            }

    float biasv[2];
#pragma unroll
    for (int nt = 0; nt < 2; ++nt) {
        const int n = nt * 16 + lo;
        biasv[nt]   = (n < HH) ? (bi[n] + bh[n]) : 0.f;
    }

    const int  gr    = bt * 16 + lo;  // A-matrix row (batch) for this lane
    const long inrow = (long)gr * TT;

    // Wait for producer's first chunk (+1 for the cross-step refill) before
    // touching any input rows. Flags only take values 0, CHK, 2*CHK, ...
    if (!first) {
        if (lane == 0) {
            const int need = (CHK + 1 < TT) ? (CHK + 1) : TT;
            while (__hip_atomic_load(srcflag, __ATOMIC_ACQUIRE,
                                     __HIP_MEMORY_SCOPE_AGENT) < need)
                __builtin_amdgcn_s_sleep(2);
        }
        __syncthreads();
    }

    // ---- software pipeline: preload input fragments for t = 0 ----
    v2f ain[8];
    {
        const float* ip = in + inrow * istr;
        if (first) {
#pragma unroll
            for (int kk = 0; kk < 4; ++kk) {
                const int k0 = 4 * kk + 2 * c;
                ain[kk][0]   = ip[k0];
                ain[kk][1]   = ip[k0 + 1];
            }
        } else {
#pragma unroll
            for (int kk = 0; kk < 8; ++kk) {
                const int k0 = 4 * kk + 2 * c;
                ain[kk][0]   = ip[k0];
                ain[kk][1]   = ip[k0 + 1];
            }
        }
    }

    for (int t = 0; t < TT; ++t) {
        // ---- chunk-granular acquire on the producer (skew = 2 chunks) ----
        if (!first && t && ((t & (CHK - 1)) == 0)) {
            if (lane == 0) {
                const int need = (t + CHK + 1 < TT) ? (t + CHK + 1) : TT;
                while (__hip_atomic_load(srcflag, __ATOMIC_ACQUIRE,
                                         __HIP_MEMORY_SCOPE_AGENT) < need)
                    __builtin_amdgcn_s_sleep(2);
            }
            __syncthreads();
        }

        // ---- xw = x_t @ Wi^T + bias (independent of the recurrence) ----
        v8f accx[2];
#pragma unroll
        for (int nt = 0; nt < 2; ++nt)
#pragma unroll
            for (int v = 0; v < 8; ++v) accx[nt][v] = biasv[nt];
        if (first) {
#pragma unroll
            for (int kk = 0; kk < 4; ++kk) {
                accx[0] = wmma_f32_k4(ain[kk], wiB[kk][0], accx[0]);
                accx[1] = wmma_f32_k4(ain[kk], wiB[kk][1], accx[1]);
            }
        } else {
#pragma unroll
            for (int kk = 0; kk < 8; ++kk) {
                accx[0] = wmma_f32_k4(ain[kk], wiB[kk][0], accx[0]);
                accx[1] = wmma_f32_k4(ain[kk], wiB[kk][1], accx[1]);
            }
        }

        // ---- refill input fragments for step t+1 (whole step to land) ----
        {
            const int    tn = (t + 1 < TT) ? (t + 1) : 0;  // branch-free guard
            const float* ip = in + (inrow + tn) * istr;
            __builtin_prefetch(ip + 2 * CHK * istr, 0, 3);
            if (first) {
#pragma unroll
                for (int kk = 0; kk < 4; ++kk) {
                    const int k0 = 4 * kk + 2 * c;
                    ain[kk][0]   = ip[k0];
                    ain[kk][1]   = ip[k0 + 1];
                }
            } else {
#pragma unroll
                for (int kk = 0; kk < 8; ++kk) {
                    const int k0 = 4 * kk + 2 * c;
                    ain[kk][0]   = ip[k0];
                    ain[kk][1]   = ip[k0 + 1];
                }
            }
        }

        // ---- + h_{t-1} @ Wh^T : two independent chains of 4 WMMAs ----
        v8f acch0[2], acch1[2];
#pragma unroll
        for (int nt = 0; nt < 2; ++nt)
#pragma unroll
            for (int v = 0; v < 8; ++v) {
                acch0[nt][v] = 0.f;
                acch1[nt][v] = 0.f;
            }
#pragma unroll
        for (int kk = 0; kk < 4; ++kk) {
            const int k0a = 4 * kk + 2 * c;        // chain 0: kk 0..3
            const int k0b = 4 * (kk + 4) + 2 * c;  // chain 1: kk 4..7
            v2f a0 = *(const v2f*)&hbuf[lo * LDSS + k0a];
            v2f a1 = *(const v2f*)&hbuf[lo * LDSS + k0b];
            acch0[0] = wmma_f32_k4(a0, whB[kk][0], acch0[0]);
            acch0[1] = wmma_f32_k4(a0, whB[kk][1], acch0[1]);
            acch1[0] = wmma_f32_k4(a1, whB[kk + 4][0], acch1[0]);
            acch1[1] = wmma_f32_k4(a1, whB[kk + 4][1], acch1[1]);
        }

        v8f acc[2];
#pragma unroll
        for (int nt = 0; nt < 2; ++nt) acc[nt] = accx[nt] + acch0[nt] + acch1[nt];

        __syncthreads();  // all reads of h_{t-1} done before overwrite

        // ---- h_t = tanh(acc), stage for next step, emit to activations ----
        const bool emit = !last || (t == TT - 1);  // layer 9: only FC row
#pragma unroll
        for (int nt = 0; nt < 2; ++nt) {
            const int  n    = nt * 16 + lo;
            const bool live = (n < HH);
#pragma unroll
            for (int v = 0; v < 8; ++v) {
                const float hv = live ? tanh_fast(acc[nt][v]) : 0.f;
                const int   m  = v + 8 * c;  // batch row in tile
                hbuf[m * LDSS + n] = hv;     // D-layout -> LDS for next step
                if (emit)
                    out[((long)(bt * 16 + m) * TT + t) * OSTR + n] = hv;
            }
        }
        __syncthreads();  // h_t visible before next step's LDS loads

        // ---- release-publish progress every CHK steps (fence amortized) ----
        if (!last && ((t & (CHK - 1)) == (CHK - 1))) {
            if (lane == 0)
                __hip_atomic_store(myflag, t + 1, __ATOMIC_RELEASE,
                                   __HIP_MEMORY_SCOPE_AGENT);
        }
    }
}

// FC on the last timestep: out[b] = h9[b][T-1][:] . fc_w + fc_b
__global__ void fc_last(const float* __restrict__ hin,
                        const float* __restrict__ fcw,
                        const float* __restrict__ fcb,
                        float* __restrict__ out) {
    const int b = blockIdx.x * blockDim.x + threadIdx.x;
    if (b >= BB) return;
    const float* row = hin + ((long)b * TT + (TT - 1)) * OSTR;
    float s = fcb[0];
#pragma unroll
    for (int n = 0; n < HH; ++n) s += row[n] * fcw[n];
    out[b] = s;
}

extern "C" void kernel_launch(void* const* d_in, const int* in_sizes, int n_in,
                              void* d_out, int out_size, void* d_ws,
                              size_t ws_size, hipStream_t stream) {
    const float* x     = (const float*)d_in[0];  // [B,T,16]
    const float* W_ih0 = (const float*)d_in[1];  // [H,16]
    const float* W_ih  = (const float*)d_in[2];  // [L-1,H,H]
    const float* W_hh  = (const float*)d_in[3];  // [L,H,H]
    const float* b_ih  = (const float*)d_in[4];  // [L,H]
    const float* b_hh  = (const float*)d_in[5];  // [L,H]
    const float* fc_w  = (const float*)d_in[6];  // [1,H]
    const float* fc_b  = (const float*)d_in[7];  // [1]

    const size_t bufElems = (size_t)BB * TT * OSTR;  // 16.7M floats = 64 MB
    float*       buf0     = (float*)d_ws;
    float*       buf1     = buf0 + bufElems;
    int*         flags    = (int*)(buf1 + bufElems);  // LL*NBT progress flags

    const int nflags = LL * NBT;
    zero_flags<<<(nflags + 255) / 256, 256, 0, stream>>>(flags, nflags);

    // One fused launch: 640 single-wave blocks, wavefront-pipelined layers.
    rnn_pipeline<<<LL * NBT, 32, 0, stream>>>(x, W_ih0, W_ih, W_hh, b_ih, b_hh,
                                              buf0, buf1, flags);

    // layer 9 (odd) wrote buf1; only row T-1 is populated/needed.
    fc_last<<<(BB + 255) / 256, 256, 0, stream>>>(buf1, fc_w, fc_b,
                                                  (float*)d_out);
}